// RbcNetwork_73684458930839
// MI455X (gfx1250) — compile-verified
//
#include <hip/hip_runtime.h>

// ---------------------------------------------------------------------------
// RBC network: 9216 batched 96x96 power iterations (50 steps), then a
// weighted, scale-normalized reduction to a 96-vector.
//
// Design for MI455X (gfx1250, wave32):
//  * one workgroup (192 threads = 6 waves) per 96x96 matrix
//  * W = sigmoid(wr*rz + rc) built once into LDS (row stride padded to 98
//    floats -> conflict-free column-strided fragment loads)
//  * matvec y = W*b done with V_WMMA_F32_16X16X4_F32: each wave owns a
//    16-row block; B-fragment is b[k..k+3] broadcast across the 16 columns
//    (A 16x4 and B 4x16 fragments both select k = 4*kk + 2*(lane>=16)+vgpr
//    per the CDNA5 VGPR layouts, so each lane loads 2 consecutive floats)
//  * L2 norm via wave32 shfl_xor butterfly, normalize, repeat 50x
//  * deterministic 2-stage output reduction through d_ws (atomic fallback
//    only if the workspace is too small)
// ---------------------------------------------------------------------------

typedef __attribute__((ext_vector_type(2))) float v2f;
typedef __attribute__((ext_vector_type(4))) float v4f;
typedef __attribute__((ext_vector_type(8))) float v8f;

#define NN      96
#define NMAT    (NN * NN)     /* 9216 matrices            */
#define MATSZ   (NN * NN)     /* 9216 elements per matrix */
#define WSTRIDE 98            /* padded LDS row stride    */
#define NITER   50
#define NTHREAD 192           /* 6 wave32 waves           */
#define EPSF    1e-12f

template <bool USE_WS>
__global__ __launch_bounds__(NTHREAD)
void rbc_power_kernel(const float* __restrict__ r_zeros,
                      const float* __restrict__ r_const,
                      const float* __restrict__ weights_r,
                      const float* __restrict__ weights_t,
                      float* __restrict__ ws_or_out)
{
    __shared__ __align__(16) float Wlds[NN * WSTRIDE];  // ~36.8 KB
    __shared__ __align__(16) float bvec[NN];
    __shared__ __align__(16) float yvec[NN];
    __shared__ float inv_norm;

    const int p   = blockIdx.x;            // matrix index = s*96 + t
    const int tid = threadIdx.x;
    const size_t base = (size_t)p * MATSZ; // contiguous 9216-float block

    // ---- Phase 1: W = sigmoid(weights_r * r_zeros + r_const) into LDS ----
    {
        const v4f* rz4 = (const v4f*)(r_zeros   + base);
        const v4f* rc4 = (const v4f*)(r_const   + base);
        const v4f* wr4 = (const v4f*)(weights_r + base);
        #pragma unroll
        for (int c = 0; c < (MATSZ / 4) / NTHREAD; ++c) {   // 12 float4 / thread
            const int f = c * NTHREAD + tid;                // float4 index
            v4f z = wr4[f] * rz4[f] + rc4[f];
            const int e = f * 4;                // 96 % 4 == 0: stays in one row
            const int i = e / NN, j = e % NN;
            float* dst = &Wlds[i * WSTRIDE + j];
            dst[0] = 1.0f / (1.0f + __expf(-z.x));
            dst[1] = 1.0f / (1.0f + __expf(-z.y));
            dst[2] = 1.0f / (1.0f + __expf(-z.z));
            dst[3] = 1.0f / (1.0f + __expf(-z.w));
        }
    }
    if (tid < NN) bvec[tid] = 1.0f;   // b0 = ones
    __syncthreads();

    // ---- Phase 2: 50 power-iteration steps, matvec via WMMA f32 16x16x4 ----
    const int lane   = tid & 31;
    const int wave   = tid >> 5;          // 0..5 -> 16-row block
    const int m0     = wave * 16;
    const int lanehi = lane >> 4;         // selects K pair per CDNA5 layout
    const int row    = m0 + (lane & 15);
    const float* Arow = &Wlds[row * WSTRIDE];

    for (int it = 0; it < NITER; ++it) {
        v8f acc = {};
        #pragma unroll
        for (int kk = 0; kk < NN / 4; ++kk) {         // 24 K-steps of 4
            const int kb = kk * 4 + lanehi * 2;
            v2f a = *(const v2f*)(Arow + kb);         // A 16x4 fragment
            v2f b = *(const v2f*)(&bvec[kb]);         // B 4x16: b broadcast over N
            acc = __builtin_amdgcn_wmma_f32_16x16x4_f32(
                false, a, false, b, (short)0, acc, false, false);
        }
        // All 16 columns of D are identical; C/D layout: VGPR v holds
        // M=v (lanes 0-15) and M=v+8 (lanes 16-31).
        if (lane == 0) {
            #pragma unroll
            for (int v = 0; v < 8; ++v) yvec[m0 + v] = acc[v];
        } else if (lane == 16) {
            #pragma unroll
            for (int v = 0; v < 8; ++v) yvec[m0 + 8 + v] = acc[v];
        }
        __syncthreads();

        // ||y|| reduction by wave 0 (wave32 butterfly)
        if (wave == 0) {
            float s = yvec[lane]      * yvec[lane]
                    + yvec[lane + 32] * yvec[lane + 32]
                    + yvec[lane + 64] * yvec[lane + 64];
            #pragma unroll
            for (int off = 16; off >= 1; off >>= 1)
                s += __shfl_xor(s, off, 32);
            if (lane == 0) inv_norm = 1.0f / (sqrtf(s) + EPSF);
        }
        __syncthreads();
        if (tid < NN) bvec[tid] = yvec[tid] * inv_norm;
        __syncthreads();
    }

    // ---- Phase 3: contribution  weights_t[s,t] * v[k] / v[s] ----
    const int s_idx   = p / NN;                 // v_ss = v[s,t,s]
    const float scale = weights_t[p] / bvec[s_idx];
    if (tid < NN) {
        const float val = bvec[tid] * scale;
        if (USE_WS) ws_or_out[(size_t)p * NN + tid] = val;
        else        atomicAdd(&ws_or_out[tid], val);
    }
}

// Deterministic reduction: block k sums partial[p][k] over p = 0..9215.
__global__ __launch_bounds__(256)
void rbc_reduce_kernel(const float* __restrict__ ws, float* __restrict__ out)
{
    __shared__ float red[256];
    const int k = blockIdx.x;
    const int t = threadIdx.x;
    float s = 0.0f;
    for (int pp = t; pp < NMAT; pp += 256)
        s += ws[(size_t)pp * NN + k];
    red[t] = s;
    __syncthreads();
    #pragma unroll
    for (int off = 128; off >= 1; off >>= 1) {
        if (t < off) red[t] += red[t + off];
        __syncthreads();
    }
    if (t == 0) out[k] = red[0];
}

__global__ void rbc_zero_kernel(float* __restrict__ out)
{
    if (threadIdx.x < NN) out[threadIdx.x] = 0.0f;
}

extern "C" void kernel_launch(void* const* d_in, const int* in_sizes, int n_in,
                              void* d_out, int out_size, void* d_ws, size_t ws_size,
                              hipStream_t stream)
{
    // setup_inputs order: 0=x (unused), 1=r_zeros, 2=r_const, 3=max_error
    // (unused), 4=weights_t, 5=weights_r
    const float* r_zeros   = (const float*)d_in[1];
    const float* r_const   = (const float*)d_in[2];
    const float* weights_t = (const float*)d_in[4];
    const float* weights_r = (const float*)d_in[5];
    float* out = (float*)d_out;

    const size_t need = (size_t)NMAT * NN * sizeof(float);  // ~3.4 MB
    if (d_ws != nullptr && ws_size >= need) {
        float* ws = (float*)d_ws;
        rbc_power_kernel<true><<<NMAT, NTHREAD, 0, stream>>>(
            r_zeros, r_const, weights_r, weights_t, ws);
        rbc_reduce_kernel<<<NN, 256, 0, stream>>>(ws, out);
    } else {
        rbc_zero_kernel<<<1, NN, 0, stream>>>(out);
        rbc_power_kernel<false><<<NMAT, NTHREAD, 0, stream>>>(
            r_zeros, r_const, weights_r, weights_t, out);
    }
}